// CombinedLoss_51943334477835
// MI455X (gfx1250) — compile-verified
//
#include <hip/hip_runtime.h>
#include <hip/hip_bf16.h>
#include <math.h>

#define T_STEPS 6
#define B_SZ    512
#define S_SZ    40
#define V_SZ    28
#define NPEAKS  512
#define NRES    38          // S-2
#define NIONS   37          // NRES-1
#define NTHEO   111         // 3*NIONS

#define PROTON_MASS 1.007276f
#define WATER_MASS  18.010565f
#define CO_MASS     27.994915f
#define MASS_TOL    0.5f
#define INV_TEMP    10.0f   // 1/0.1
#define HUB_DELTA   0.2f

typedef __attribute__((ext_vector_type(2))) float v2f;
typedef __attribute__((ext_vector_type(8))) float v8f;

__device__ __forceinline__ float wave_sum(float v) {
    for (int off = 16; off > 0; off >>= 1) v += __shfl_xor(v, off, 32);
    return v;
}
__device__ __forceinline__ float wave_min(float v) {
    for (int off = 16; off > 0; off >>= 1) v = fminf(v, __shfl_xor(v, off, 32));
    return v;
}

// ws[0]=mask_sum, ws[1]=ce_acc, ws[2]=spec_num, ws[3]=spec_cnt
__global__ void k_init_masksum(const unsigned char* __restrict__ tmask, float* __restrict__ acc) {
    __shared__ float sred[256];
    int tid = threadIdx.x;
    float s = 0.f;
    for (int i = tid; i < B_SZ * S_SZ; i += 256) s += tmask[i] ? 1.f : 0.f;
    sred[tid] = s;
    __syncthreads();
    for (int st = 128; st > 0; st >>= 1) {
        if (tid < st) sred[tid] += sred[tid + st];
        __syncthreads();
    }
    if (tid == 0) { acc[0] = sred[0]; acc[1] = 0.f; acc[2] = 0.f; acc[3] = 0.f; }
}

// one block per (t,b); thread s handles one V=28 row
__global__ void k_ce(const float* __restrict__ logits, const int* __restrict__ targets,
                     const unsigned char* __restrict__ tmask, float* __restrict__ acc) {
    int tb = blockIdx.x;
    int t = tb / B_SZ, b = tb % B_SZ;
    int s = threadIdx.x;
    float val = 0.f;
    if (s < S_SZ) {
        int tgt = targets[b * S_SZ + s];
        float mf = tmask[b * S_SZ + s] ? 1.f : 0.f;
        if (tgt != 0 && mf != 0.f) {
            const float* row = logits + (((size_t)t * B_SZ + b) * S_SZ + s) * V_SZ;
            float m = -INFINITY;
            #pragma unroll
            for (int v = 0; v < V_SZ; ++v) m = fmaxf(m, row[v]);
            float sum = 0.f;
            #pragma unroll
            for (int v = 0; v < V_SZ; ++v) sum += __expf(row[v] - m);
            float ce = __logf(sum) + m - row[tgt];
            val = ((float)(t + 1) * (1.0f / 21.0f)) * ce * mf;
        }
    }
    __shared__ float sred[64];
    sred[threadIdx.x] = val;
    __syncthreads();
    for (int st = 32; st > 0; st >>= 1) {
        if (threadIdx.x < st) sred[threadIdx.x] += sred[threadIdx.x + st];
        __syncthreads();
    }
    if (threadIdx.x == 0) atomicAdd(&acc[1], sred[0]);
}

// one wave per 16 flattened (b,s) rows: softmax -> LDS -> 7x WMMA f32 16x16x4
__global__ void __launch_bounds__(32) k_expected_wmma(const float* __restrict__ last_logits,
                                                      const float* __restrict__ aa,
                                                      float* __restrict__ expected) {
    __shared__ float probs[16][32];
    int lane = threadIdx.x;
    int base = blockIdx.x * 16;
    if (lane < 16) {
        const float* rp = last_logits + (size_t)(base + lane) * V_SZ;
        float m = -INFINITY;
        float e[V_SZ];
        #pragma unroll
        for (int v = 0; v < V_SZ; ++v) m = fmaxf(m, rp[v]);
        float sum = 0.f;
        #pragma unroll
        for (int v = 0; v < V_SZ; ++v) { e[v] = __expf(rp[v] - m); sum += e[v]; }
        float inv = 1.f / sum;
        #pragma unroll
        for (int v = 0; v < V_SZ; ++v) probs[lane][v] = e[v] * inv;
        for (int v = V_SZ; v < 32; ++v) probs[lane][v] = 0.f;
    }
    __syncthreads();

    // A 16x4 f32 layout: lanes 0-15 -> M=lane, K=0(v0),1(v1); lanes 16-31 -> K=2,3.
    // B 4x16 f32 layout mirrors A with lane = N; replicate aa across all N.
    int m16  = lane & 15;
    int koff = (lane < 16) ? 0 : 2;
    v8f c = {};
    #pragma unroll
    for (int kk = 0; kk < 7; ++kk) {   // 7 * K4 = K28 exactly
        v2f a, bb;
        a.x  = probs[m16][kk * 4 + koff];
        a.y  = probs[m16][kk * 4 + koff + 1];
        bb.x = aa[kk * 4 + koff];
        bb.y = aa[kk * 4 + koff + 1];
        c = __builtin_amdgcn_wmma_f32_16x16x4_f32(false, a, false, bb, (short)0, c, false, false);
    }
    // D layout: VGPR r, lanes 0-15: M=r,N=lane ; lanes 16-31: M=8+r,N=lane-16.
    // Column N=0 lives in lanes 0 and 16.
    if (lane == 0) {
        #pragma unroll
        for (int r = 0; r < 8; ++r) expected[base + r] = c[r];
    } else if (lane == 16) {
        #pragma unroll
        for (int r = 0; r < 8; ++r) expected[base + 8 + r] = c[r];
    }
}

// b/y/a ion ladders: serial 38-element prefix/suffix per b (trivial work)
__global__ void k_ions(const float* __restrict__ expected, float* __restrict__ theo) {
    int b = blockIdx.x;
    if (threadIdx.x != 0) return;
    const float* e = expected + b * S_SZ;
    float res[NRES];
    #pragma unroll
    for (int i = 0; i < NRES; ++i) res[i] = e[i + 1];
    float* tb = theo + b * NTHEO;
    float acc = 0.f;
    for (int i = 0; i < NIONS; ++i) {
        acc += res[i];
        tb[i]             = acc + PROTON_MASS;             // b-ion
        tb[2 * NIONS + i] = acc + PROTON_MASS - CO_MASS;   // a-ion
    }
    float sacc = 0.f;
    for (int j = NRES - 1; j >= 0; --j) {
        sacc += res[j];
        if (j < NIONS) tb[NIONS + j] = sacc + WATER_MASS + PROTON_MASS;  // y-ion
    }
}

// One block per b: async-stage 4.6KB of peak data into LDS once, then
// 8 waves sweep the 111 ions against LDS-resident peaks.
__global__ void __launch_bounds__(256) k_spec(const float* __restrict__ theo,
                                              const float* __restrict__ om,
                                              const float* __restrict__ oi,
                                              const unsigned char* __restrict__ pmask,
                                              float* __restrict__ acc) {
    __shared__ float lom[NPEAKS];
    __shared__ float loi[NPEAKS];
    __shared__ unsigned char lpm[NPEAKS];
    __shared__ float snum, scnt;

    int b    = blockIdx.x;
    int tid  = threadIdx.x;
    int lane = tid & 31;
    int wid  = tid >> 5;

    if (tid == 0) { snum = 0.f; scnt = 0.f; }

    if (tid < 32) {
        // Async global->LDS: 32 lanes x 16B = 512B per issue (ASYNCcnt path).
        // Generic shared pointers truncate to the LDS byte address (ISA: LDS aperture uses addr[31:0]).
        unsigned long long gom = (unsigned long long)(uintptr_t)(om + (size_t)b * NPEAKS);
        unsigned long long goi = (unsigned long long)(uintptr_t)(oi + (size_t)b * NPEAKS);
        unsigned long long gpm = (unsigned long long)(uintptr_t)(pmask + (size_t)b * NPEAKS);
        unsigned lom0 = (unsigned)(uintptr_t)&lom[0];
        unsigned loi0 = (unsigned)(uintptr_t)&loi[0];
        unsigned lpm0 = (unsigned)(uintptr_t)&lpm[0];
        #pragma unroll
        for (int it = 0; it < 4; ++it) {                 // 4 x 512B = 2KB each
            unsigned      rel = (unsigned)(lane * 16 + it * 512);
            unsigned long long ga = gom + rel;
            unsigned           la = lom0 + rel;
            asm volatile("global_load_async_to_lds_b128 %0, %1, off"
                         :: "v"(la), "v"(ga) : "memory");
            unsigned long long gb = goi + rel;
            unsigned           lb = loi0 + rel;
            asm volatile("global_load_async_to_lds_b128 %0, %1, off"
                         :: "v"(lb), "v"(gb) : "memory");
        }
        {                                                // 1 x 512B mask
            unsigned           rel = (unsigned)(lane * 16);
            unsigned long long gc  = gpm + rel;
            unsigned           lc  = lpm0 + rel;
            asm volatile("global_load_async_to_lds_b128 %0, %1, off"
                         :: "v"(lc), "v"(gc) : "memory");
        }
        asm volatile("s_wait_asynccnt 0x0" ::: "memory");
    }
    __syncthreads();

    for (int k = wid; k < NTHEO; k += 8) {               // one wave per ion
        float tv = theo[b * NTHEO + k];

        float dmin = 1e30f;
        for (int j = lane; j < NPEAKS; j += 32) {
            if (lpm[j]) {
                float d = fabsf(tv - lom[j]);
                if (d < MASS_TOL) dmin = fminf(dmin, d);
            }
        }
        dmin = wave_min(dmin);
        if (dmin <= MASS_TOL) {                          // wave-uniform predicate
            float Z = 0.f, Hs = 0.f, Is = 0.f;
            for (int j = lane; j < NPEAKS; j += 32) {
                if (lpm[j]) {
                    float d = fabsf(tv - lom[j]);
                    if (d < MASS_TOL) {
                        float e   = __expf((dmin - d) * INV_TEMP);
                        float hub = (d <= HUB_DELTA) ? 0.5f * d * d
                                                     : HUB_DELTA * (d - 0.5f * HUB_DELTA);
                        Z  += e;
                        Hs += e * hub;
                        Is += e * loi[j];
                    }
                }
            }
            Z  = wave_sum(Z);
            Hs = wave_sum(Hs);
            Is = wave_sum(Is);
            if (lane == 0) {
                float invZ = 1.f / Z;
                atomicAdd(&snum, (Hs * invZ) * (Is * invZ));
                atomicAdd(&scnt, 1.0f);
            }
        }
    }
    __syncthreads();
    if (tid == 0) {
        atomicAdd(&acc[2], snum);
        atomicAdd(&acc[3], scnt);
    }
}

__global__ void k_final(const float* __restrict__ acc, float* __restrict__ out) {
    if (threadIdx.x == 0 && blockIdx.x == 0) {
        float denom = fmaxf(acc[0], 1.0f);
        float ce    = acc[1] / denom;
        float cnt   = fmaxf(acc[3], 1.0f);
        float spec  = acc[2] / cnt;
        out[0] = 1.0f * ce + 0.1f * spec;
    }
}

extern "C" void kernel_launch(void* const* d_in, const int* in_sizes, int n_in,
                              void* d_out, int out_size, void* d_ws, size_t ws_size,
                              hipStream_t stream) {
    const float*         all_logits = (const float*)d_in[0];
    const int*           targets    = (const int*)d_in[1];
    const unsigned char* tmask      = (const unsigned char*)d_in[2];
    const float*         om         = (const float*)d_in[3];
    const float*         oi         = (const float*)d_in[4];
    const unsigned char* pmask      = (const unsigned char*)d_in[5];
    const float*         aa         = (const float*)d_in[6];

    float* acc      = (float*)d_ws;
    float* expected = acc + 16;                  // B*S floats
    float* theo     = expected + B_SZ * S_SZ;    // B*111 floats
    float* out      = (float*)d_out;

    k_init_masksum<<<1, 256, 0, stream>>>(tmask, acc);
    k_ce<<<T_STEPS * B_SZ, 64, 0, stream>>>(all_logits, targets, tmask, acc);

    const float* last = all_logits + (size_t)(T_STEPS - 1) * B_SZ * S_SZ * V_SZ;
    k_expected_wmma<<<(B_SZ * S_SZ) / 16, 32, 0, stream>>>(last, aa, expected);

    k_ions<<<B_SZ, 32, 0, stream>>>(expected, theo);

    k_spec<<<B_SZ, 256, 0, stream>>>(theo, om, oi, pmask, acc);

    k_final<<<1, 1, 0, stream>>>(acc, out);
}